// VGNN_76527727280738
// MI455X (gfx1250) — compile-verified
//
#include <hip/hip_runtime.h>

// GraphSAGE 2-layer + linear head for MI455X (gfx1250, wave32).
// Phase costs: edge scatter dominates but is L2-resident (192MB L2 >> 90MB
// working set); GEMMs are ~5 GFLOP total -> use full-precision f32 WMMA
// (V_WMMA_F32_16X16X4_F32) rather than a low-precision path.

#define IN_CH 64
#define HIDDEN 128

typedef __attribute__((ext_vector_type(2))) float v2f;
typedef __attribute__((ext_vector_type(8))) float v8f;

__device__ __forceinline__ v8f wmma4(v2f a, v2f b, v8f c) {
  // 8 args: (neg_a, A, neg_b, B, c_mod, C, reuse_a, reuse_b)
  return __builtin_amdgcn_wmma_f32_16x16x4_f32(false, a, false, b, (short)0, c,
                                               false, false);
}

__global__ void init_out_kernel(float* __restrict__ out,
                                const float* __restrict__ b_lin, int n) {
  int i = blockIdx.x * blockDim.x + threadIdx.x;
  if (i < n) out[i] = b_lin[0];
}

// One thread per (edge, feature). Consecutive threads share an edge and walk
// consecutive features -> coalesced loads from feat[src] and coalesced
// global_atomic_add_f32 bursts into agg[dst] (L2-resident accumulators).
template <int LOGF, bool COUNT_DEG>
__global__ void scatter_kernel(const float* __restrict__ feat,
                               const int* __restrict__ src,
                               const int* __restrict__ dst,
                               float* __restrict__ agg,
                               float* __restrict__ deg, int nE) {
  unsigned tid = blockIdx.x * blockDim.x + threadIdx.x;
  unsigned e = tid >> LOGF;
  unsigned f = tid & ((1u << LOGF) - 1u);
  if (e >= (unsigned)nE) return;
  int s = src[e];
  int d = dst[e];
  unsafeAtomicAdd(&agg[((size_t)d << LOGF) + f], feat[((size_t)s << LOGF) + f]);
  if (COUNT_DEG && f == 0) unsafeAtomicAdd(&deg[d], 1.0f);
}

// SAGE layer GEMM: out = relu( (agg/deg) @ Wl + bias + xin @ Wr ).
// One 16x16 output tile per wave (V_WMMA_F32_16X16X4_F32); 8 waves per block
// cover the 128 output columns; gridDim.x covers rows in exact 16-row tiles
// (50000 = 16*3125, so EXEC stays all-1s as WMMA requires).
// FINAL=true fuses the head: row_out = relu(tile) @ Wlin accumulated into
// out_vec (pre-initialized with b_lin), skipping the h2 store entirely.
template <int K, bool FINAL>
__global__ void sage_gemm_kernel(const float* __restrict__ agg,
                                 const float* __restrict__ deg,
                                 const float* __restrict__ xin,
                                 const float* __restrict__ Wl,
                                 const float* __restrict__ Wr,
                                 const float* __restrict__ bias,
                                 float* __restrict__ out_h,
                                 const float* __restrict__ Wlin,
                                 float* __restrict__ out_vec) {
  const int lane = threadIdx.x & 31;
  const int wave = threadIdx.x >> 5;
  const int half = lane >> 4;  // selects K sub-pair for A/B operands
  const int l16 = lane & 15;
  const int m = blockIdx.x * 16 + l16;  // A-matrix row held by this lane
  const int n = wave * 16 + l16;        // B/C/D column held by this lane

  const float rdeg = 1.0f / fmaxf(deg[m], 1.0f);

  v8f acc;
  const float bv = bias[n];
#pragma unroll
  for (int v = 0; v < 8; ++v) acc[v] = bv;

  const float* aggRow = agg + (size_t)m * K;
  const float* xRow = xin + (size_t)m * K;

#pragma unroll 4
  for (int k0 = 0; k0 < K; k0 += 4) {
    const int ka = k0 + half * 2;  // ISA layout: lanes 0-15 K={0,1}, 16-31 K={2,3}
    v2f a1, a2, b1, b2;
    a1[0] = aggRow[ka] * rdeg;
    a1[1] = aggRow[ka + 1] * rdeg;
    a2[0] = xRow[ka];
    a2[1] = xRow[ka + 1];
    b1[0] = Wl[ka * HIDDEN + n];
    b1[1] = Wl[(ka + 1) * HIDDEN + n];
    b2[0] = Wr[ka * HIDDEN + n];
    b2[1] = Wr[(ka + 1) * HIDDEN + n];
    acc = wmma4(a1, b1, acc);  // mean @ Wl
    acc = wmma4(a2, b2, acc);  // x    @ Wr
  }

  if (!FINAL) {
#pragma unroll
    for (int v = 0; v < 8; ++v) {
      // C/D layout: VGPR v holds M = v (lanes 0-15) or v+8 (lanes 16-31)
      int mr = blockIdx.x * 16 + half * 8 + v;
      out_h[(size_t)mr * HIDDEN + n] = fmaxf(acc[v], 0.0f);
    }
  } else {
    const float wl = Wlin[n];
#pragma unroll
    for (int v = 0; v < 8; ++v) {
      float p = fmaxf(acc[v], 0.0f) * wl;
      // reduce the 16 column-lanes of this half-wave (wave32; masks < 16
      // keep the two halves independent)
      p += __shfl_xor(p, 1, 32);
      p += __shfl_xor(p, 2, 32);
      p += __shfl_xor(p, 4, 32);
      p += __shfl_xor(p, 8, 32);
      if (l16 == 0) {
        int mr = blockIdx.x * 16 + half * 8 + v;
        unsafeAtomicAdd(&out_vec[mr], p);  // 8 waves contribute partials
      }
    }
  }
}

extern "C" void kernel_launch(void* const* d_in, const int* in_sizes, int n_in,
                              void* d_out, int out_size, void* d_ws,
                              size_t ws_size, hipStream_t stream) {
  const float* x = (const float*)d_in[0];
  const int* ei = (const int*)d_in[1];
  const float* W1l = (const float*)d_in[2];
  const float* b1 = (const float*)d_in[3];
  const float* W1r = (const float*)d_in[4];
  const float* W2l = (const float*)d_in[5];
  const float* b2 = (const float*)d_in[6];
  const float* W2r = (const float*)d_in[7];
  const float* Wlin = (const float*)d_in[8];
  const float* blin = (const float*)d_in[9];
  float* out = (float*)d_out;

  const int N = in_sizes[0] / IN_CH;  // 50000 (multiple of 16)
  const int nE = in_sizes[1] / 2;     // 800000
  const int* src = ei;
  const int* dst = ei + nE;

  // Workspace layout (~64.2 MB): deg | agg1[N,64] | h1[N,128] | agg2[N,128]
  auto align512 = [](size_t v) { return (v + 511) & ~(size_t)511; };
  char* ws = (char*)d_ws;
  size_t o = 0;
  float* deg = (float*)(ws + o);
  o = align512(o + (size_t)N * 4);
  float* agg1 = (float*)(ws + o);
  o = align512(o + (size_t)N * IN_CH * 4);
  float* h1 = (float*)(ws + o);
  o = align512(o + (size_t)N * HIDDEN * 4);
  float* agg2 = (float*)(ws + o);
  o = align512(o + (size_t)N * HIDDEN * 4);
  (void)ws_size;
  (void)n_in;
  (void)out_size;

  hipMemsetAsync(deg, 0, (size_t)N * 4, stream);
  hipMemsetAsync(agg1, 0, (size_t)N * IN_CH * 4, stream);
  hipMemsetAsync(agg2, 0, (size_t)N * HIDDEN * 4, stream);
  init_out_kernel<<<(N + 255) / 256, 256, 0, stream>>>(out, blin, N);

  {  // layer-1 neighbor sum + degree
    long long t = (long long)nE << 6;
    scatter_kernel<6, true><<<(unsigned)((t + 255) / 256), 256, 0, stream>>>(
        x, src, dst, agg1, deg, nE);
  }
  sage_gemm_kernel<IN_CH, false><<<N / 16, 256, 0, stream>>>(
      agg1, deg, x, W1l, W1r, b1, h1, nullptr, nullptr);
  {  // layer-2 neighbor sum
    long long t = (long long)nE << 7;
    scatter_kernel<7, false><<<(unsigned)((t + 255) / 256), 256, 0, stream>>>(
        h1, src, dst, agg2, nullptr, nE);
  }
  sage_gemm_kernel<HIDDEN, true><<<N / 16, 256, 0, stream>>>(
      agg2, deg, h1, W2l, W2r, b2, nullptr, Wlin, out);
}